// LeakyZOPlainOnce_11287174053917
// MI455X (gfx1250) — compile-verified
//
#include <hip/hip_runtime.h>
#include <stdint.h>

// Leaky integrate-and-fire over T=32 steps with LocalZO STE (forward pass).
// Memory-bound streaming scan: 402 MB of HBM traffic, ~17us floor at 23.3 TB/s.
// CDNA5 path used: async global->LDS (ASYNCcnt) double-buffered staging,
// wave-private (no barriers), NT stores for the write-once output.

typedef float v4f __attribute__((ext_vector_type(4)));

#define T_STEPS 32
#define BLOCK   256   // 8 wave32s per block

// Low 32 bits of a flat "shared" pointer == LDS byte offset (aperture layout).
__device__ __forceinline__ uint32_t lds_off_u32(const void* p) {
  return (uint32_t)(uintptr_t)p;
}

// GLOBAL_LOAD_ASYNC_TO_LDS_B128 (GVS mode): copies from global base saddr plus offset vaddr to the LDS address in vdst,
// 16 bytes per lane, tracked by ASYNCcnt (per-wave).
__device__ __forceinline__ void async_load_b128(uint32_t lds_byte,
                                                uint32_t goff_byte,
                                                uint64_t base) {
  asm volatile("global_load_async_to_lds_b128 %0, %1, %2"
               :
               : "v"(lds_byte), "v"(goff_byte), "s"(base)
               : "memory");
}

__global__ __launch_bounds__(BLOCK) void lif_localzo_scan(
    const float* __restrict__ x,   // inputs          [T, n] flattened
    const float* __restrict__ z,   // random tangents [T, n] flattened
    float* __restrict__ out,       // spikes          [T, n] flattened
    unsigned n)                    // elements per time step (B*F)
{
  __shared__ v4f xbuf[2][BLOCK];   // 2 * 4KB
  __shared__ v4f zbuf[2][BLOCK];   // 2 * 4KB

  const unsigned tid = threadIdx.x;
  const unsigned e0  = (blockIdx.x * BLOCK + tid) * 4u;  // first element of quad
  if (e0 >= n) return;                                   // whole-wave granular here

  const uint64_t xb    = (uint64_t)x;
  const uint64_t zb    = (uint64_t)z;
  const unsigned stepB = n * 4u;                         // bytes per time step
  unsigned goff        = e0 * 4u;                        // byte offset in step image

  const uint32_t lx0 = lds_off_u32(&xbuf[0][tid]);
  const uint32_t lx1 = lds_off_u32(&xbuf[1][tid]);
  const uint32_t lz0 = lds_off_u32(&zbuf[0][tid]);
  const uint32_t lz1 = lds_off_u32(&zbuf[1][tid]);

  // Prologue: stage t = 0.
  async_load_b128(lx0, goff, xb);
  async_load_b128(lz0, goff, zb);

  v4f u;
  u.x = 0.0f; u.y = 0.0f; u.z = 0.0f; u.w = 0.0f;

  for (int t = 0; t < T_STEPS; ++t) {
    const int cur = t & 1;

    if (t + 1 < T_STEPS) {
      // Prefetch t+1 into the other buffer, then drain down to the 2 newest
      // outstanding async ops (async loads complete in order) so that the
      // pair for step t is guaranteed landed in LDS.
      const unsigned goff_n = goff + stepB;
      async_load_b128(cur ? lx0 : lx1, goff_n, xb);
      async_load_b128(cur ? lz0 : lz1, goff_n, zb);
      asm volatile("s_wait_asynccnt 0x2" ::: "memory");
    } else {
      asm volatile("s_wait_asynccnt 0x0" ::: "memory");
    }

    // Each lane reads exactly the LDS slot its own async loads wrote:
    // wave-private pipeline, no workgroup barrier needed.
    v4f xv = xbuf[cur][tid];
    v4f zv = zbuf[cur][tid];

    v4f sp;
#pragma unroll
    for (int c = 0; c < 4; ++c) {
      float uu = 0.5f * u[c] + xv[c];          // membrane update (BETA = 0.5)
      float v  = uu - 1.0f;                    // distance to threshold (U_TH = 1)
      float az = __builtin_fabsf(zv[c]);
      // LocalZO surrogate: |z|/(2*delta) * 1(|v| < delta*|z|), delta = 0.05
      float surr = (__builtin_fabsf(v) < 0.05f * az) ? (az / 0.1f) : 0.0f;
      float h    = (v > 0.0f) ? 1.0f : 0.0f;   // hard spike
      float sv   = surr * v;
      float spike = (h - sv) + sv;             // straight-through forward value
      u[c]  = uu - spike;                      // soft reset (subtract U_TH)
      sp[c] = spike;
    }

    // Write-once output: non-temporal store, don't pollute near caches.
    __builtin_nontemporal_store(sp, (v4f*)(out + (size_t)t * n + e0));

    goff += stepB;
  }
}

extern "C" void kernel_launch(void* const* d_in, const int* in_sizes, int n_in,
                              void* d_out, int out_size, void* d_ws, size_t ws_size,
                              hipStream_t stream) {
  (void)n_in; (void)out_size; (void)d_ws; (void)ws_size;

  const float* x   = (const float*)d_in[0];   // inputs          (fp32)
  const float* z   = (const float*)d_in[1];   // random_tangents (fp32)
  float*       out = (float*)d_out;           // spikes          (fp32)

  const long long total = in_sizes[0];        // T * B * F
  const unsigned  n     = (unsigned)(total / T_STEPS);   // B * F per step

  const unsigned elems_per_block = BLOCK * 4u;           // 1024 elements
  const unsigned blocks = (n + elems_per_block - 1) / elems_per_block;

  lif_localzo_scan<<<blocks, BLOCK, 0, stream>>>(x, z, out, n);
}